// DanceDynamicsModel_1709396984196
// MI455X (gfx1250) — compile-verified
//
#include <hip/hip_runtime.h>
#include <math.h>

#define DN 128
#define NQ 7

typedef __attribute__((ext_vector_type(2))) float v2f;
typedef __attribute__((ext_vector_type(8))) float v8f;

// ---- workspace layout (float offsets) ----
constexpr int OFF_AR    = 0;          // real(A) = -0.5*M
constexpr int OFF_AI    = 16384;      // imag(A) = -H
constexpr int OFF_AIN   = 32768;      // -imag(A) = +H  (avoids WMMA negation modifiers)
constexpr int OFF_RR    = 49152;      // rho real plane
constexpr int OFF_RI    = 65536;      // rho imag plane
constexpr int OFF_YR    = 81920;      // RK4 stage input real
constexpr int OFF_YI    = 98304;      // RK4 stage input imag
constexpr int OFF_K     = 114688;     // k1R,k1I,k2R,k2I,k3R,k3I,k4R,k4I (8 planes)
constexpr int OFF_SMALL = 245760;     // small precomputed operators
// small-region sub-offsets (relative to OFF_SMALL)
constexpr int S_OPS = 0;     // 7 * 4   : MLP 2x2 ops
constexpr int S_LS  = 32;    // 7 * 4   : single-site Lindblad 2x2
constexpr int S_LP  = 64;    // 42 * 16 : pair Lindblad 4x4
constexpr int S_HIS = 736;   // 7 * 4   : single-site Hamiltonian pieces
constexpr int S_HIJ = 768;   // 21 * 16 : pair Hamiltonian pieces
constexpr int S_MIS = 1104;  // 7 * 4   : single-site Gram (L^T L)
constexpr int S_MPS = 1136;  // 42 * 16 : pair Gram

// ============================================================================
// Kernel 1: tiny prologue — MLP -> ops, then all small (2x2 / 4x4) operators.
// ============================================================================
__global__ void prep_small(const float* __restrict__ feat,
                           const float* __restrict__ W1, const float* __restrict__ b1,
                           const float* __restrict__ W2, const float* __restrict__ b2,
                           const float* __restrict__ Hs, const float* __restrict__ Hc,
                           const float* __restrict__ rates, float* __restrict__ S) {
  int t = threadIdx.x;
  if (t < NQ) {
    float f0 = feat[t * 2 + 0], f1 = feat[t * 2 + 1];
    float o[4];
    for (int k = 0; k < 4; ++k) o[k] = b2[k];
    for (int h = 0; h < 64; ++h) {
      float a = f0 * W1[h] + f1 * W1[64 + h] + b1[h];
      a = a > 0.f ? a : 0.f;
      for (int k = 0; k < 4; ++k) o[k] += a * W2[h * 4 + k];
    }
    for (int k = 0; k < 4; ++k) S[S_OPS + t * 4 + k] = o[k];
  }
  __syncthreads();
  if (t < NQ) {
    const float* op = &S[S_OPS + t * 4];
    const float* hs = &Hs[t * 4];
    float ls[4], his[4];
    for (int a = 0; a < 2; ++a)
      for (int b = 0; b < 2; ++b) {
        float g = sqrtf(fabsf(rates[((t * NQ + t) * 4 + a) * 4 + b]));
        ls[a * 2 + b] = g * op[a * 2 + b];
        float s1 = 0.f, s2 = 0.f;
        for (int c = 0; c < 2; ++c) {
          s1 += op[a * 2 + c] * hs[c * 2 + b];   // op @ Hs
          s2 += hs[c * 2 + a] * op[b * 2 + c];   // Hs^T @ op^T
        }
        his[a * 2 + b] = s1 + s2;
      }
    for (int k = 0; k < 4; ++k) { S[S_LS + t * 4 + k] = ls[k]; S[S_HIS + t * 4 + k] = his[k]; }
    for (int a = 0; a < 2; ++a)
      for (int b = 0; b < 2; ++b) {
        float s = 0.f;
        for (int c = 0; c < 2; ++c) s += ls[c * 2 + a] * ls[c * 2 + b];  // L^T L
        S[S_MIS + t * 4 + a * 2 + b] = s;
      }
  }
  if (t < 42) {  // ordered pairs i != j
    int i = t / 6, jr = t % 6, j = jr + (jr >= i ? 1 : 0);
    const float* oi = &S[S_OPS + i * 4];
    const float* oj = &S[S_OPS + j * 4];
    float lp[16];
    for (int ai = 0; ai < 2; ++ai)
      for (int aj = 0; aj < 2; ++aj)
        for (int bi = 0; bi < 2; ++bi)
          for (int bj = 0; bj < 2; ++bj) {
            int a = ai * 2 + aj, b = bi * 2 + bj;
            float g = sqrtf(fabsf(rates[((i * NQ + j) * 4 + a) * 4 + b]));
            lp[a * 4 + b] = g * oi[ai * 2 + bi] * oj[aj * 2 + bj];
          }
    for (int k = 0; k < 16; ++k) S[S_LP + t * 16 + k] = lp[k];
    for (int a = 0; a < 4; ++a)
      for (int b = 0; b < 4; ++b) {
        float s = 0.f;
        for (int c = 0; c < 4; ++c) s += lp[c * 4 + a] * lp[c * 4 + b];
        S[S_MPS + t * 16 + a * 4 + b] = s;
      }
  }
  if (t < 21) {  // unordered pairs i<j for Hamiltonian coupling
    int q = t, i = 0, rem = q;
    while (rem >= (NQ - 1 - i)) { rem -= (NQ - 1 - i); ++i; }
    int j = i + 1 + rem;
    const float* oi = &S[S_OPS + i * 4];
    const float* oj = &S[S_OPS + j * 4];
    const float* hc = &Hc[(i * NQ + j) * 16];
    float K4[16];
    for (int ai = 0; ai < 2; ++ai)
      for (int aj = 0; aj < 2; ++aj)
        for (int bi = 0; bi < 2; ++bi)
          for (int bj = 0; bj < 2; ++bj)
            K4[(ai * 2 + aj) * 4 + bi * 2 + bj] = oi[ai * 2 + bi] * oj[aj * 2 + bj];
    for (int a = 0; a < 4; ++a)
      for (int b = 0; b < 4; ++b) {
        float s1 = 0.f, s2 = 0.f;
        for (int c = 0; c < 4; ++c) {
          s1 += K4[a * 4 + c] * hc[c * 4 + b];
          s2 += hc[c * 4 + a] * K4[b * 4 + c];
        }
        S[S_HIJ + q * 16 + a * 4 + b] = s1 + s2;
      }
  }
}

// ============================================================================
// Kernel 2: assemble dense A = -iH - 0.5M via embedding deltas.
// qubit i <-> bit (6-i) of the 7-bit index.
// ============================================================================
__global__ void build_A(const float* __restrict__ S, float* __restrict__ Ar,
                        float* __restrict__ Ai, float* __restrict__ Ain) {
  int idx = blockIdx.x * blockDim.x + threadIdx.x;
  int r = idx >> 7, c = idx & 127;
  int x = r ^ c;
  float H = 0.f, M = 0.f;
  for (int i = 0; i < NQ; ++i) {
    int sh = 6 - i;
    if ((x & ~(1 << sh)) == 0) {
      int br = (r >> sh) & 1, bc = (c >> sh) & 1;
      H += S[S_HIS + i * 4 + br * 2 + bc];
      M += S[S_MIS + i * 4 + br * 2 + bc];
    }
  }
  int q = 0;
  for (int i = 0; i < NQ; ++i)
    for (int j = i + 1; j < NQ; ++j, ++q) {
      int shi = 6 - i, shj = 6 - j;
      int msk = (1 << shi) | (1 << shj);
      if ((x & ~msk) == 0) {
        int rs = ((r >> shi) & 1) * 2 + ((r >> shj) & 1);
        int cs = ((c >> shi) & 1) * 2 + ((c >> shj) & 1);
        H += S[S_HIJ + q * 16 + rs * 4 + cs];
      }
    }
  for (int p = 0; p < 42; ++p) {
    int i = p / 6, jr = p % 6, j = jr + (jr >= i ? 1 : 0);
    int shi = 6 - i, shj = 6 - j;
    int msk = (1 << shi) | (1 << shj);
    if ((x & ~msk) == 0) {
      int rs = ((r >> shi) & 1) * 2 + ((r >> shj) & 1);
      int cs = ((c >> shi) & 1) * 2 + ((c >> shj) & 1);
      M += S[S_MPS + p * 16 + rs * 4 + cs];
    }
  }
  Ar[idx] = -0.5f * M;
  Ai[idx] = -H;
  Ain[idx] = H;
}

__global__ void init_rho(const float* __restrict__ rho0, float* __restrict__ rr,
                         float* __restrict__ ri, float* __restrict__ out) {
  int idx = blockIdx.x * blockDim.x + threadIdx.x;
  float v = rho0[idx];
  rr[idx] = v; ri[idx] = 0.f;
  out[2 * idx] = v; out[2 * idx + 1] = 0.f;
}

// ============================================================================
// Complex 16x16 tile of X = A @ P via v_wmma_f32_16x16x4_f32 (wave32).
// A from global (L2-resident planes); B (P) from LDS.
// ============================================================================
__device__ __forceinline__ void ctile(const float* __restrict__ Ar, const float* __restrict__ Ai,
                                      const float* __restrict__ Ain,
                                      const float* sPr, const float* sPi,
                                      int RB, int CB, int lane, v8f& accR, v8f& accI) {
  int col = lane & 15;
  int koff = (lane >> 4) * 2;
  int arow = (RB + col) * DN;
  for (int kb = 0; kb < DN; kb += 4) {
    int k0 = kb + koff;
    v2f aR, aI, aN, bR, bI;
    aR.x = Ar[arow + k0];  aR.y = Ar[arow + k0 + 1];
    aI.x = Ai[arow + k0];  aI.y = Ai[arow + k0 + 1];
    aN.x = Ain[arow + k0]; aN.y = Ain[arow + k0 + 1];
    bR.x = sPr[k0 * DN + CB + col]; bR.y = sPr[(k0 + 1) * DN + CB + col];
    bI.x = sPi[k0 * DN + CB + col]; bI.y = sPi[(k0 + 1) * DN + CB + col];
    // Xr += Ar*Pr - Ai*Pi ; Xi += Ar*Pi + Ai*Pr
    accR = __builtin_amdgcn_wmma_f32_16x16x4_f32(false, aR, false, bR, (short)0, accR, false, false);
    accR = __builtin_amdgcn_wmma_f32_16x16x4_f32(false, aN, false, bI, (short)0, accR, false, false);
    accI = __builtin_amdgcn_wmma_f32_16x16x4_f32(false, aR, false, bI, (short)0, accI, false, false);
    accI = __builtin_amdgcn_wmma_f32_16x16x4_f32(false, aI, false, bR, (short)0, accI, false, false);
  }
}

// ============================================================================
// Kernel 3: one full RHS eval.  K = A@P + (A@P)^dagger + Sum_k L_k P L_k^dagger.
// One wave per block; block owns tile (I,J) + computes (J,I) for the dagger.
// Both P planes (128KB) are staged into LDS (320KB/WGP on CDNA5) with
// GLOBAL_LOAD_ASYNC_TO_LDS_B128 (ASYNCcnt path) and serve the WMMA B-frags
// and the scattered dissipator reads.
// ============================================================================
__global__ void __launch_bounds__(32) rhs_kernel(const float* __restrict__ Ar,
                                                 const float* __restrict__ Ai,
                                                 const float* __restrict__ Ain,
                                                 const float* __restrict__ Pr,
                                                 const float* __restrict__ Pi,
                                                 const float* __restrict__ S,
                                                 float* __restrict__ Kr,
                                                 float* __restrict__ Ki) {
  __shared__ __align__(16) float sPr[DN * DN];
  __shared__ __align__(16) float sPi[DN * DN];
  __shared__ float sLs[28];
  __shared__ float sLp[672];
  __shared__ float xR[256], xI[256];
  int l = threadIdx.x;

  // ---- async-stage both P planes into LDS (lane-cyclic 16B chunks) ----
  {
    // low 32 bits of a generic pointer to LDS == LDS byte address (ISA 10.2)
    unsigned ldsR = (unsigned)(size_t)(void*)sPr;
    unsigned ldsI = (unsigned)(size_t)(void*)sPi;
    for (int n = l; n < (DN * DN) / 4; n += 32) {
      unsigned dR = ldsR + (unsigned)n * 16u;
      unsigned dI = ldsI + (unsigned)n * 16u;
      unsigned long long gR = (unsigned long long)(const void*)(Pr + n * 4);
      unsigned long long gI = (unsigned long long)(const void*)(Pi + n * 4);
      asm volatile("global_load_async_to_lds_b128 %0, %1, off" :: "v"(dR), "v"(gR) : "memory");
      asm volatile("global_load_async_to_lds_b128 %0, %1, off" :: "v"(dI), "v"(gI) : "memory");
    }
  }
  for (int n = l; n < 28; n += 32)  sLs[n] = S[S_LS + n];
  for (int n = l; n < 672; n += 32) sLp[n] = S[S_LP + n];
  asm volatile("s_wait_asynccnt 0x0" ::: "memory");
  __syncthreads();

  int I = blockIdx.x >> 3, J = blockIdx.x & 7;
  int col = l & 15, half = l >> 4;
  int RB = I * 16, CB = J * 16;
  int cg = CB + col;  // this lane's fixed global column

  v8f aR1 = {}, aI1 = {}, aR2 = {}, aI2 = {};
  ctile(Ar, Ai, Ain, sPr, sPi, RB, CB, l, aR1, aI1);   // X(I,J)
  ctile(Ar, Ai, Ain, sPr, sPi, CB, RB, l, aR2, aI2);   // X(J,I)

  // stash X(J,I) tile to LDS; C-layout: elem(m,n) m=v+8*half, n=col
#pragma unroll
  for (int v = 0; v < 8; ++v) {
    int m = v + 8 * half;
    xR[m * 16 + col] = aR2[v];
    xI[m * 16 + col] = aI2[v];
  }
  __syncthreads();

  // ---- dissipator: column side hoisted per lane (c fixed), rows per element
  float dR[8], dI[8];
#pragma unroll
  for (int v = 0; v < 8; ++v) { dR[v] = 0.f; dI[v] = 0.f; }

  // single-site Lindblad terms (real 2x2 ops)
  for (int i = 0; i < NQ; ++i) {
    int sh = 6 - i;
    int bc = (cg >> sh) & 1;
    int cb = cg & ~(1 << sh);
    float w0 = sLs[i * 4 + bc * 2 + 0], w1 = sLs[i * 4 + bc * 2 + 1];
    int cc0 = cb, cc1 = cb | (1 << sh);
#pragma unroll
    for (int v = 0; v < 8; ++v) {
      int r = RB + v + 8 * half;
      int br = (r >> sh) & 1;
      int rb = r & ~(1 << sh);
      for (int b = 0; b < 2; ++b) {
        int rr = (rb | (b << sh)) * DN;
        float wr = sLs[i * 4 + br * 2 + b];
        dR[v] += wr * (w0 * sPr[rr + cc0] + w1 * sPr[rr + cc1]);
        dI[v] += wr * (w0 * sPi[rr + cc0] + w1 * sPi[rr + cc1]);
      }
    }
  }
  // pairwise Lindblad terms (real 4x4 ops on qubit bits (i,j))
  for (int p = 0; p < 42; ++p) {
    int i = p / 6, jr = p % 6, j = jr + (jr >= i ? 1 : 0);
    int shi = 6 - i, shj = 6 - j;
    int cs = ((cg >> shi) & 1) * 2 + ((cg >> shj) & 1);
    int msk = ~((1 << shi) | (1 << shj));
    int cb = cg & msk;
    float wc[4]; int cc[4];
#pragma unroll
    for (int b2 = 0; b2 < 4; ++b2) {
      wc[b2] = sLp[p * 16 + cs * 4 + b2];
      cc[b2] = cb | (((b2 >> 1) & 1) << shi) | ((b2 & 1) << shj);
    }
#pragma unroll
    for (int v = 0; v < 8; ++v) {
      int r = RB + v + 8 * half;
      int rs = ((r >> shi) & 1) * 2 + ((r >> shj) & 1);
      int rb = r & msk;
      for (int b = 0; b < 4; ++b) {
        int rr = (rb | (((b >> 1) & 1) << shi) | ((b & 1) << shj)) * DN;
        float wr = sLp[p * 16 + rs * 4 + b];
        float sR = 0.f, sI = 0.f;
#pragma unroll
        for (int b2 = 0; b2 < 4; ++b2) {
          sR += wc[b2] * sPr[rr + cc[b2]];
          sI += wc[b2] * sPi[rr + cc[b2]];
        }
        dR[v] += wr * sR;
        dI[v] += wr * sI;
      }
    }
  }

#pragma unroll
  for (int v = 0; v < 8; ++v) {
    int m = v + 8 * half;
    int r = RB + m, c = cg;
    float tR = aR1[v] + xR[col * 16 + m];   // X + X^dagger (real)
    float tI = aI1[v] - xI[col * 16 + m];   // X + X^dagger (imag)
    Kr[r * DN + c] = tR + dR[v];
    Ki[r * DN + c] = tI + dI[v];
  }
}

// ============================================================================
// RK4 glue kernels (dt read from t_eval on device)
// ============================================================================
__global__ void stage_axpy(const float* __restrict__ rr, const float* __restrict__ ri,
                           const float* __restrict__ kr, const float* __restrict__ ki,
                           float* __restrict__ yr, float* __restrict__ yi,
                           const float* __restrict__ te, int step, float f) {
  int idx = blockIdx.x * blockDim.x + threadIdx.x;
  float co = f * (te[step + 1] - te[step]);
  yr[idx] = rr[idx] + co * kr[idx];
  yi[idx] = ri[idx] + co * ki[idx];
}

__global__ void rk4_final(float* __restrict__ rr, float* __restrict__ ri,
                          const float* __restrict__ K, const float* __restrict__ te,
                          int step, float* __restrict__ out) {
  int idx = blockIdx.x * blockDim.x + threadIdx.x;
  float s = (te[step + 1] - te[step]) * (1.f / 6.f);
  const float *k1r = K, *k1i = K + 16384, *k2r = K + 2 * 16384, *k2i = K + 3 * 16384,
              *k3r = K + 4 * 16384, *k3i = K + 5 * 16384, *k4r = K + 6 * 16384, *k4i = K + 7 * 16384;
  float nr = rr[idx] + s * (k1r[idx] + 2.f * k2r[idx] + 2.f * k3r[idx] + k4r[idx]);
  float ni = ri[idx] + s * (k1i[idx] + 2.f * k2i[idx] + 2.f * k3i[idx] + k4i[idx]);
  rr[idx] = nr; ri[idx] = ni;
  float* o = out + ((size_t)(step + 1) * 16384 + idx) * 2;
  o[0] = nr; o[1] = ni;
}

extern "C" void kernel_launch(void* const* d_in, const int* in_sizes, int n_in,
                              void* d_out, int out_size, void* d_ws, size_t ws_size,
                              hipStream_t stream) {
  const float* feat  = (const float*)d_in[0];
  const float* te    = (const float*)d_in[1];
  const float* W1    = (const float*)d_in[2];
  const float* b1    = (const float*)d_in[3];
  const float* W2    = (const float*)d_in[4];
  const float* b2    = (const float*)d_in[5];
  const float* Hs    = (const float*)d_in[6];
  const float* Hc    = (const float*)d_in[7];
  const float* rates = (const float*)d_in[8];
  const float* rho0  = (const float*)d_in[9];
  float* out = (float*)d_out;

  float* W = (float*)d_ws;
  float *Ar = W + OFF_AR, *Ai = W + OFF_AI, *Ain = W + OFF_AIN;
  float *rr = W + OFF_RR, *ri = W + OFF_RI, *yr = W + OFF_YR, *yi = W + OFF_YI;
  float *K = W + OFF_K, *S = W + OFF_SMALL;
  float *k1r = K, *k1i = K + 16384, *k2r = K + 2 * 16384, *k2i = K + 3 * 16384,
        *k3r = K + 4 * 16384, *k3i = K + 5 * 16384, *k4r = K + 6 * 16384, *k4i = K + 7 * 16384;

  prep_small<<<1, 64, 0, stream>>>(feat, W1, b1, W2, b2, Hs, Hc, rates, S);
  build_A<<<128, 128, 0, stream>>>(S, Ar, Ai, Ain);
  init_rho<<<128, 128, 0, stream>>>(rho0, rr, ri, out);

  int steps = in_sizes[1] - 1;  // T-1 RK4 steps
  for (int step = 0; step < steps; ++step) {
    rhs_kernel<<<64, 32, 0, stream>>>(Ar, Ai, Ain, rr, ri, S, k1r, k1i);
    stage_axpy<<<128, 128, 0, stream>>>(rr, ri, k1r, k1i, yr, yi, te, step, 0.5f);
    rhs_kernel<<<64, 32, 0, stream>>>(Ar, Ai, Ain, yr, yi, S, k2r, k2i);
    stage_axpy<<<128, 128, 0, stream>>>(rr, ri, k2r, k2i, yr, yi, te, step, 0.5f);
    rhs_kernel<<<64, 32, 0, stream>>>(Ar, Ai, Ain, yr, yi, S, k3r, k3i);
    stage_axpy<<<128, 128, 0, stream>>>(rr, ri, k3r, k3i, yr, yi, te, step, 1.0f);
    rhs_kernel<<<64, 32, 0, stream>>>(Ar, Ai, Ain, yr, yi, S, k4r, k4i);
    rk4_final<<<128, 128, 0, stream>>>(rr, ri, K, te, step, out);
  }
}